// HighPassImageLoss_17583596110202
// MI455X (gfx1250) — compile-verified
//
#include <hip/hip_runtime.h>
#include <hip/hip_bf16.h>
#include <stdint.h>

// Problem constants (from reference): B=8, T=24, H=W=320
#define NFRAMES 192
#define HH      320
#define WW      320                // 320 floats = 1280 B per row (16B multiple)
#define ROWS    16                 // output rows per block tile
#define TILES   (HH / ROWS)        // 20
#define BLOCK   320                // 10 wave32s; thread == column
#define LROWS   (ROWS + 2)         // 18 staged rows (with vertical halo)
// LDS tile is contiguous rows of exactly WW floats (TDM writes tiles densely)

#define AS1 __attribute__((address_space(1)))
#define AS3 __attribute__((address_space(3)))

typedef __attribute__((ext_vector_type(4))) int          v4i;
typedef __attribute__((ext_vector_type(4))) unsigned int v4u;
typedef __attribute__((ext_vector_type(8))) int          v8i;

#if __has_builtin(__builtin_amdgcn_tensor_load_to_lds)
#define HAVE_TDM 1
#else
#define HAVE_TDM 0
#endif

#if HAVE_TDM
// One TDM descriptor moves a dense 2D tile (rows x WW fp32, row stride WW) from
// global memory into LDS at lds_byte. Issued per-wave; EXEC ignored by TDM.
__device__ __forceinline__ void tdm_load_tile(const float* gsrc, uint32_t lds_byte,
                                              int rows) {
    uint64_t ga = (uint64_t)(uintptr_t)gsrc;
    v4u g0;
    g0.x = 1u;                                              // count=1, user mode
    g0.y = lds_byte;                                        // lds_addr (bytes)
    g0.z = (uint32_t)(ga & 0xFFFFFFFFu);                    // global_addr[31:0]
    g0.w = (uint32_t)((ga >> 32) & 0x01FFFFFFu) | (2u << 30); // addr[56:32]|type=2
    v8i g1;
    g1[0] = (int)(2u << 16);                                // data_size=2 (4 bytes)
    g1[1] = (int)((uint32_t)WW << 16);                      // tensor_dim0[15:0]=320
    g1[2] = (int)((uint32_t)rows << 16);                    // tensor_dim1[15:0]=rows
    g1[3] = (int)((uint32_t)WW << 16);                      // tile_dim0=320
    g1[4] = (int)(uint32_t)rows;                            // tile_dim1=rows; tile_dim2=0
    g1[5] = WW;                                             // tensor_dim0_stride=320
    g1[6] = 0;
    g1[7] = 0;
    v4i g2 = {0, 0, 0, 0};                                  // 2D: groups 2/3 unused
    v4i g3 = {0, 0, 0, 0};
    v8i g4 = {0, 0, 0, 0, 0, 0, 0, 0};                      // extra group (clang-23 form)
    __builtin_amdgcn_tensor_load_to_lds(g0, g1, g2, g3, g4, /*cpol=*/0);
}

__device__ __forceinline__ void wait_tensor0() {
    __builtin_amdgcn_s_wait_tensorcnt(0);
}
#else
// Fallback: per-lane async global->LDS b128 staging (ASYNCcnt path).
__device__ __forceinline__ void async_copy_b128(const float* gsrc,
                                                AS3 char* lds_base,
                                                uint32_t lds_byte_off) {
#if __has_builtin(__builtin_amdgcn_global_load_async_to_lds_b128)
    __builtin_amdgcn_global_load_async_to_lds_b128(
        (AS1 v4i*)(uintptr_t)gsrc,
        (AS3 v4i*)(lds_base + lds_byte_off),
        /*imm offset=*/0, /*cpol=*/0);
#else
    uint64_t ga = (uint64_t)(uintptr_t)gsrc;
    uint32_t la = (uint32_t)(uintptr_t)(lds_base + lds_byte_off);
    asm volatile("global_load_async_to_lds_b128 %0, %1, off"
                 :: "v"(la), "v"(ga) : "memory");
#endif
}
__device__ __forceinline__ void wait_async0() {
#if __has_builtin(__builtin_amdgcn_s_wait_asynccnt)
    __builtin_amdgcn_s_wait_asynccnt(0);
#else
    asm volatile("s_wait_asynccnt 0" ::: "memory");
#endif
}
#endif

__global__ void __launch_bounds__(BLOCK)
hp_l1_kernel(const float* __restrict__ X, const float* __restrict__ Y,
             float* __restrict__ out) {
    __shared__ float tX[LROWS * WW];
    __shared__ float tY[LROWS * WW];
    __shared__ float red[BLOCK / 32];

    const int tid   = threadIdx.x;
    const int blk   = blockIdx.x;
    const int frame = blk / TILES;
    const int tile  = blk - frame * TILES;
    const int r0    = tile * ROWS;
    const bool top  = (tile == 0);
    const bool bot  = (tile == TILES - 1);

    // LDS row k <-> frame row r0-1+k, k = 0..17; valid frame rows get DMA'd,
    // the out-of-frame top/bottom halo rows are zero-filled with DS stores.
    if (top) { tX[tid] = 0.f;  tY[tid] = 0.f; }                             // row -1
    if (bot) { tX[(LROWS - 1) * WW + tid] = 0.f;
               tY[(LROWS - 1) * WW + tid] = 0.f; }                          // row 320

    const int kstart = top ? 1 : 0;
    const int nrows  = LROWS - (top ? 1 : 0) - (bot ? 1 : 0);
    const int fr0    = r0 - 1 + kstart;
    const float* Xf  = X + (size_t)frame * HH * WW;
    const float* Yf  = Y + (size_t)frame * HH * WW;

    const uint32_t ldsX_off = (uint32_t)(uintptr_t)&tX[0];
    const uint32_t ldsY_off = (uint32_t)(uintptr_t)&tY[0];

#if HAVE_TDM
    if (tid < 32) {   // wave 0 issues both tile DMAs; TENSORcnt is per-wave
        tdm_load_tile(Xf + (size_t)fr0 * WW, ldsX_off + (uint32_t)(kstart * WW * 4), nrows);
        tdm_load_tile(Yf + (size_t)fr0 * WW, ldsY_off + (uint32_t)(kstart * WW * 4), nrows);
        wait_tensor0();
    }
#else
    {
        AS3 char* ldsX = (AS3 char*)ldsX_off;
        AS3 char* ldsY = (AS3 char*)ldsY_off;
        const int G = WW / 4;                               // 80 b128 groups per row
        for (int idx = tid; idx < nrows * G; idx += BLOCK) {
            int rr = idx / G;
            int cc = (idx - rr * G) * 4;
            int k  = kstart + rr;
            int fr = fr0 + rr;
            uint32_t loff = (uint32_t)((k * WW + cc) * sizeof(float));
            async_copy_b128(Xf + (size_t)fr * WW + cc, ldsX, loff);
            async_copy_b128(Yf + (size_t)fr * WW + cc, ldsY, loff);
        }
        wait_async0();
    }
#endif
    __syncthreads();

    // ---- 9-point high-pass on D = X - Y with rolling row-sums ----
    // hp = 8*d - ring = 9*d - (3x3 sum). Left/right zero-padding handled by
    // per-thread constant masks instead of halo columns (rows are dense).
    const int   c     = tid;                 // output column 0..319
    const bool  has_l = (c > 0);
    const bool  has_r = (c < WW - 1);
    const float maskL = has_l ? 1.f : 0.f;
    const float maskR = has_r ? 1.f : 0.f;

    auto rowsum = [&](int k, float& center) -> float {
        const int base = k * WW + c;
        const int jl   = has_l ? base - 1 : base;   // clamped (value masked out)
        const int jr   = has_r ? base + 1 : base;
        float dm = tX[base] - tY[base];
        float dl = (tX[jl] - tY[jl]) * maskL;
        float dr = (tX[jr] - tY[jr]) * maskR;
        center = dm;
        return dl + dm + dr;
    };

    float acc = 0.f;
    float c_dummy, cB;
    float rsA = rowsum(0, c_dummy);
    float rsB = rowsum(1, cB);
    for (int i = 1; i <= ROWS; ++i) {
        float cC;
        float rsC = rowsum(i + 1, cC);
        float hp  = 9.f * cB - (rsA + rsB + rsC);
        acc += fabsf(hp);
        rsA = rsB; rsB = rsC; cB = cC;
    }

    // ---- wave32 shuffle reduce, then cross-wave via LDS, one atomic ----
    #pragma unroll
    for (int off = 16; off > 0; off >>= 1)
        acc += __shfl_down(acc, off, 32);
    if ((tid & 31) == 0) red[tid >> 5] = acc;
    __syncthreads();
    if (tid == 0) {
        float s = 0.f;
        #pragma unroll
        for (int w = 0; w < BLOCK / 32; ++w) s += red[w];
        const float inv_total = 1.0f / ((float)NFRAMES * (float)HH * (float)WW);
        atomicAdd(out, s * inv_total);
    }
}

__global__ void zero_out_kernel(float* out) { out[0] = 0.f; }

extern "C" void kernel_launch(void* const* d_in, const int* in_sizes, int n_in,
                              void* d_out, int out_size, void* d_ws, size_t ws_size,
                              hipStream_t stream) {
    (void)in_sizes; (void)n_in; (void)out_size; (void)d_ws; (void)ws_size;
    const float* X = (const float*)d_in[0];
    const float* Y = (const float*)d_in[1];
    float* out = (float*)d_out;

    zero_out_kernel<<<1, 1, 0, stream>>>(out);
    hp_l1_kernel<<<NFRAMES * TILES, BLOCK, 0, stream>>>(X, Y, out);
}